// GCNet_4767413698801
// MI455X (gfx1250) — compile-verified
//
#include <hip/hip_runtime.h>
#include <hip/hip_bf16.h>

// GCNet block on MI455X (gfx1250, wave32).
// x:(32,512,64,64) f32. HBM-bound (~1 GB traffic, ~0.35 GFLOP -> ~45us @ 23.3TB/s).
// Bottleneck GEMMs use V_WMMA_F32_16X16X4_F32 (fp32 matrix core path).

#define cN 32
#define cC 512
#define cH 64
#define cW 64
#define cHW (cH * cW)        // 4096
#define cCB 32               // C / 16
#define cCHUNK 4             // C-split for logits partial sums (parallelism)

typedef __attribute__((ext_vector_type(2))) float v2f;
typedef __attribute__((ext_vector_type(8))) float v8f;

// ---------------------------------------------------------------------------
// Kernel 1: partial logits. chunk q sums channels [q*128, q*128+128):
//   part[q][n,h,w] = sum_{c in chunk} x[n,c,h,w]*wk_w[c]
// One thread per 4 consecutive w positions (float4, fully coalesced).
// 4 chunks x 128 blocks = 512 workgroups to keep the whole chip streaming.
// ---------------------------------------------------------------------------
__global__ __launch_bounds__(256) void k_logits_part(const float* __restrict__ x,
                                                     const float* __restrict__ wk_w,
                                                     float* __restrict__ part) {
  int tid = blockIdx.x * blockDim.x + threadIdx.x;   // 0 .. 4*N*H*W/4-1
  int chunk = tid >> 15;                             // N*H*W/4 = 32768 per chunk
  int pos = tid & 32767;
  int n = pos >> 10;                                 // H*W/4 = 1024
  int hw4 = pos & 1023;
  int c0 = chunk * (cC / cCHUNK);
  const float4* xp = reinterpret_cast<const float4*>(
                         x + ((size_t)n * cC + c0) * cHW) + hw4;
  float4 acc = {0.f, 0.f, 0.f, 0.f};
#pragma unroll 4
  for (int c = 0; c < cC / cCHUNK; ++c) {
    float wv = wk_w[c0 + c];                         // uniform -> scalar load
    float4 xv = xp[c * (cHW / 4)];
    acc.x += xv.x * wv; acc.y += xv.y * wv;
    acc.z += xv.z * wv; acc.w += xv.w * wv;
  }
  reinterpret_cast<float4*>(part)[chunk * 32768 + pos] = acc;
}

// ---------------------------------------------------------------------------
// Kernel 2: sum partials + bias, then softmax over H (axis=1 of (n,h,w)).
// One 64-thread block per (n,w); thread = h.
// ---------------------------------------------------------------------------
__global__ __launch_bounds__(64) void k_softmax_h(const float* __restrict__ part,
                                                  const float* __restrict__ wk_b,
                                                  float* __restrict__ attn) {
  __shared__ float sm[64];
  int n = blockIdx.x >> 6;       // / W
  int w = blockIdx.x & 63;       // % W
  int h = threadIdx.x;
  int idx = (n * cH + h) * cW + w;
  float v = wk_b[0];
#pragma unroll
  for (int q = 0; q < cCHUNK; ++q) v += part[q * (cN * cHW) + idx];
  sm[h] = v;
  __syncthreads();
#pragma unroll
  for (int s = 32; s > 0; s >>= 1) {
    if (h < s) sm[h] = fmaxf(sm[h], sm[h + s]);
    __syncthreads();
  }
  float m = sm[0];
  __syncthreads();
  float e = __expf(v - m);
  sm[h] = e;
  __syncthreads();
#pragma unroll
  for (int s = 32; s > 0; s >>= 1) {
    if (h < s) sm[h] += sm[h + s];
    __syncthreads();
  }
  attn[idx] = e / sm[0];
}

// ---------------------------------------------------------------------------
// Kernel 3: ctx[n,c] = sum_hw x[n,c,hw]*attn[n,hw]. Block per (n,c),
// float4 loads, LDS tree reduction. attn (16KB/n) stays L2-hot.
// ---------------------------------------------------------------------------
__global__ __launch_bounds__(256) void k_ctx(const float* __restrict__ x,
                                             const float* __restrict__ attn,
                                             float* __restrict__ ctx) {
  __shared__ float red[256];
  int n = blockIdx.x >> 9;       // / C
  int c = blockIdx.x & 511;      // % C
  const float4* xp = reinterpret_cast<const float4*>(x + ((size_t)n * cC + c) * cHW);
  const float4* ap = reinterpret_cast<const float4*>(attn + (size_t)n * cHW);
  float acc = 0.f;
#pragma unroll
  for (int i = threadIdx.x; i < cHW / 4; i += 256) {
    float4 xv = xp[i];
    float4 av = ap[i];
    acc += xv.x * av.x + xv.y * av.y + xv.z * av.z + xv.w * av.w;
  }
  red[threadIdx.x] = acc;
  __syncthreads();
#pragma unroll
  for (int s = 128; s > 0; s >>= 1) {
    if (threadIdx.x < s) red[threadIdx.x] += red[threadIdx.x + s];
    __syncthreads();
  }
  if (threadIdx.x == 0) ctx[blockIdx.x] = red[0];
}

// ---------------------------------------------------------------------------
// Kernel 4: bottleneck, single block, 4 waves, fp32 WMMA (16x16x4).
//   v = ctx(32x512) @ wv1^T(512x32)   -> LN over 32 ch -> ReLU
//   out_vec = v(32x32) @ wv2^T(32x512)
// Lane layout (32-bit WMMA): A: lane=(m%16)+16*khalf holds K=2*khalf+{0,1};
// B symmetric with n; D: VGPR r <-> M = r + 8*khalf, N = lane%16.
// ---------------------------------------------------------------------------
__global__ __launch_bounds__(128) void k_bottleneck(const float* __restrict__ ctx,
                                                    const float* __restrict__ wv1,
                                                    const float* __restrict__ ln_g,
                                                    const float* __restrict__ ln_b,
                                                    const float* __restrict__ wv2,
                                                    float* __restrict__ out_vec) {
  __shared__ float vmat[32][32];
  const int tid = threadIdx.x;
  const int wid = tid >> 5;
  const int lane = tid & 31;
  const int row = lane & 15;     // M index for A, N index for B
  const int kh = lane >> 4;      // K-half selector

  // ---- Stage 1: v = ctx @ wv1^T   (M=32, N=32, K=512); 1 tile per wave ----
  {
    const int m0 = (wid >> 1) * 16;
    const int n0 = (wid & 1) * 16;
    v8f acc = {0.f, 0.f, 0.f, 0.f, 0.f, 0.f, 0.f, 0.f};
    for (int kb = 0; kb < cC; kb += 4) {
      const int k = kb + kh * 2;
      v2f a, b;
      a.x = ctx[(m0 + row) * cC + k];
      a.y = ctx[(m0 + row) * cC + k + 1];
      b.x = wv1[(n0 + row) * cC + k];
      b.y = wv1[(n0 + row) * cC + k + 1];
      acc = __builtin_amdgcn_wmma_f32_16x16x4_f32(false, a, false, b,
                                                  (short)0, acc, false, false);
    }
#pragma unroll
    for (int r = 0; r < 8; ++r)
      vmat[m0 + r + 8 * kh][n0 + row] = acc[r];
  }
  __syncthreads();

  // ---- LayerNorm over 32 channels per row (row = batch n) + ReLU ----
  if (tid < 32) {
    float mu = 0.f;
#pragma unroll
    for (int j = 0; j < 32; ++j) mu += vmat[tid][j];
    mu *= (1.f / 32.f);
    float var = 0.f;
#pragma unroll
    for (int j = 0; j < 32; ++j) {
      float d = vmat[tid][j] - mu;
      var += d * d;
    }
    var *= (1.f / 32.f);
    float rs = rsqrtf(var + 1e-5f);
#pragma unroll
    for (int j = 0; j < 32; ++j) {
      float t = (vmat[tid][j] - mu) * rs * ln_g[j] + ln_b[j];
      vmat[tid][j] = fmaxf(t, 0.f);
    }
  }
  __syncthreads();

  // ---- Stage 2: out_vec = v @ wv2^T (M=32, N=512, K=32); 64 tiles ----
  for (int t = wid; t < 64; t += 4) {          // uniform per wave
    const int m0 = (t >> 5) * 16;
    const int n0 = (t & 31) * 16;
    v8f acc = {0.f, 0.f, 0.f, 0.f, 0.f, 0.f, 0.f, 0.f};
#pragma unroll
    for (int kb = 0; kb < 32; kb += 4) {
      const int k = kb + kh * 2;
      v2f a, b;
      a.x = vmat[m0 + row][k];
      a.y = vmat[m0 + row][k + 1];
      b.x = wv2[(n0 + row) * cCB + k];
      b.y = wv2[(n0 + row) * cCB + k + 1];
      acc = __builtin_amdgcn_wmma_f32_16x16x4_f32(false, a, false, b,
                                                  (short)0, acc, false, false);
    }
#pragma unroll
    for (int r = 0; r < 8; ++r)
      out_vec[(m0 + r + 8 * kh) * cC + n0 + row] = acc[r];
  }
}

// ---------------------------------------------------------------------------
// Kernel 5: out[n,c,h,w] = x[n,c,h,w] + out_vec[n,c]. float4, coalesced.
// ---------------------------------------------------------------------------
__global__ __launch_bounds__(256) void k_add(const float4* __restrict__ x4,
                                             const float* __restrict__ out_vec,
                                             float4* __restrict__ out4) {
  size_t i = (size_t)blockIdx.x * blockDim.x + threadIdx.x;   // N*C*HW/4 threads
  float s = out_vec[i >> 10];                                 // HW/4 = 1024 per (n,c)
  float4 v = x4[i];
  v.x += s; v.y += s; v.z += s; v.w += s;
  out4[i] = v;
}

// ---------------------------------------------------------------------------
extern "C" void kernel_launch(void* const* d_in, const int* in_sizes, int n_in,
                              void* d_out, int out_size, void* d_ws, size_t ws_size,
                              hipStream_t stream) {
  const float* x    = (const float*)d_in[0];   // (N,C,H,W)
  const float* wk_w = (const float*)d_in[1];   // (C,)
  const float* wk_b = (const float*)d_in[2];   // (1,)
  const float* wv1  = (const float*)d_in[3];   // (CB,C)
  const float* ln_g = (const float*)d_in[4];   // (CB,)
  const float* ln_b = (const float*)d_in[5];   // (CB,)
  const float* wv2  = (const float*)d_in[6];   // (C,CB)
  float* out = (float*)d_out;                  // (N,C,H,W)

  float* ws      = (float*)d_ws;
  float* part    = ws;                                   // 4*N*H*W = 524288
  float* attn    = part + (size_t)cCHUNK * cN * cHW;     // N*H*W   = 131072
  float* ctx     = attn + (size_t)cN * cHW;              // N*C     = 16384
  float* out_vec = ctx + (size_t)cN * cC;                // N*C     = 16384

  // 1) partial logits: 4 chunks * 32768 threads = 512 workgroups
  k_logits_part<<<(cCHUNK * cN * cHW / 4) / 256, 256, 0, stream>>>(x, wk_w, part);

  // 2) combine + softmax over H: one 64-thread block per (n,w)
  k_softmax_h<<<cN * cW, 64, 0, stream>>>(part, wk_b, attn);

  // 3) context pooling: block per (n,c)
  k_ctx<<<cN * cC, 256, 0, stream>>>(x, attn, ctx);

  // 4) bottleneck (WMMA): single block, 4 waves
  k_bottleneck<<<1, 128, 0, stream>>>(ctx, wv1, ln_g, ln_b, wv2, out_vec);

  // 5) broadcast add: N*C*H*W/4 = 16,777,216 threads
  k_add<<<(size_t)cN * cC * cHW / 4 / 256, 256, 0, stream>>>(
      (const float4*)x, out_vec, (float4*)out);
}